// mymodel_39067022524703
// MI455X (gfx1250) — compile-verified
//
#include <hip/hip_runtime.h>
#include <math.h>

typedef float v2f __attribute__((ext_vector_type(2)));
typedef float v8f __attribute__((ext_vector_type(8)));

#define DD 20
#define WPB 8            // waves per block (wave32)
#define BLOCK 256
#define GRID 1024
#define LSTR 34          // LDS row stride in floats (conflict-avoiding)

// D = A(16x4,f32) * B(4x16,f32) + C(16x16,f32), full fp32 matrix pipe.
static __device__ __forceinline__ v8f wmma4(v2f a, v2f b, v8f c) {
    return __builtin_amdgcn_wmma_f32_16x16x4_f32(
        /*neg_a=*/false, a, /*neg_b=*/false, b,
        /*c_mod=*/(short)0, c, /*reuse_a=*/false, /*reuse_b=*/false);
}

__global__ void __launch_bounds__(BLOCK)
mlp_chain_reduce(const float* __restrict__ X, const float* __restrict__ W,
                 const float* __restrict__ bvec, const float* __restrict__ Wi,
                 float* __restrict__ acc, int num_tiles) {
    __shared__ float lds[WPB][16 * LSTR];
    __shared__ float wred[WPB][2];

    const int tid  = threadIdx.x;
    const int wave = tid >> 5;          // wave32
    const int lane = tid & 31;
    const int n16  = lane & 15;
    const int koff = (lane < 16) ? 0 : 2;
    const int rbase = (lane < 16) ? 0 : 8;

    // ---- hoisted B-fragments (constant across all row tiles) ----
    // matmul 1 & 3: B[k,n] = W[n,k]  -> fragment = row n of W
    // matmul 2    : B[k,n] = Wi[k,n] -> column-strided gather
    v2f bw[2][5], bwi[2][5];
    float bias[2];
    for (int nt = 0; nt < 2; ++nt) {
        const int n = nt * 16 + n16;
        const bool nv = (n < DD);
        bias[nt] = nv ? bvec[n] : 0.0f;
        for (int s = 0; s < 5; ++s) {
            const int k = 4 * s + koff;
            v2f f = {0.0f, 0.0f};
            if (nv) f = *(const v2f*)(W + n * DD + k);      // W[n,k], W[n,k+1]
            bw[nt][s] = f;
            v2f g = {0.0f, 0.0f};
            if (nv) { g.x = Wi[k * DD + n]; g.y = Wi[(k + 1) * DD + n]; }
            bwi[nt][s] = g;
        }
    }

    float accS = 0.0f, accA = 0.0f;
    float* H = &lds[wave][0];
    const int stride = gridDim.x * WPB;

    for (int t = blockIdx.x * WPB + wave; t < num_tiles; t += stride) {
        const float* xrow = X + (size_t)(t * 16 + n16) * DD + koff;
        __builtin_prefetch(X + (size_t)((t + stride) * 16 + n16) * DD, 0, 0);

        // ---- matmul 1: H1 = X @ W^T (+b) ----
        v8f c0 = {}, c1 = {};
        #pragma unroll
        for (int s = 0; s < 5; ++s) {
            v2f a = *(const v2f*)(xrow + 4 * s);
            c0 = wmma4(a, bw[0][s], c0);
            c1 = wmma4(a, bw[1][s], c1);
        }
        #pragma unroll
        for (int j = 0; j < 8; ++j) {
            H[(rbase + j) * LSTR + n16]      = c0[j] + bias[0];
            H[(rbase + j) * LSTR + 16 + n16] = c1[j] + bias[1];
        }
        asm volatile("s_wait_dscnt 0x0" ::: "memory");  // same-wave LDS RAW

        // ---- matmul 2: H2 = relu(H1 @ Wi + 1) ----
        const float* hr = H + n16 * LSTR + koff;
        v8f d0 = {}, d1 = {};
        #pragma unroll
        for (int s = 0; s < 5; ++s) {
            v2f a = *(const v2f*)(hr + 4 * s);
            d0 = wmma4(a, bwi[0][s], d0);
            d1 = wmma4(a, bwi[1][s], d1);
        }
        asm volatile("s_wait_dscnt 0x0" ::: "memory");  // WAR: reads before overwrite
        #pragma unroll
        for (int j = 0; j < 8; ++j) {
            float v0 = d0[j] + 1.0f; v0 = v0 > 0.0f ? v0 : 0.0f;
            float v1 = d1[j] + 1.0f; v1 = v1 > 0.0f ? v1 : 0.0f;
            H[(rbase + j) * LSTR + n16]      = v0;
            H[(rbase + j) * LSTR + 16 + n16] = v1;
        }
        asm volatile("s_wait_dscnt 0x0" ::: "memory");

        // ---- matmul 3: H3 = H2 @ W^T (+b) ----
        v8f e0 = {}, e1 = {};
        #pragma unroll
        for (int s = 0; s < 5; ++s) {
            v2f a = *(const v2f*)(hr + 4 * s);
            e0 = wmma4(a, bw[0][s], e0);
            e1 = wmma4(a, bw[1][s], e1);
        }
        asm volatile("" ::: "memory");  // keep next iter's stores after these reads

        // ---- partial reduction: cols 0..15 valid; cols 16..19 only if n16<4 ----
        float sl = 0.0f, al = 0.0f;
        #pragma unroll
        for (int j = 0; j < 8; ++j) {
            float v0 = e0[j] + bias[0];
            sl += v0; al += fabsf(v0);
        }
        if (n16 < 4) {
            #pragma unroll
            for (int j = 0; j < 8; ++j) {
                float v1 = e1[j] + bias[1];
                sl += v1; al += fabsf(v1);
            }
        }
        accS += sl; accA += al;
    }

    // wave32 butterfly reduce
    #pragma unroll
    for (int off = 16; off >= 1; off >>= 1) {
        accS += __shfl_xor(accS, off, 32);
        accA += __shfl_xor(accA, off, 32);
    }
    if (lane == 0) { wred[wave][0] = accS; wred[wave][1] = accA; }
    __syncthreads();
    if (tid == 0) {
        float S = 0.0f, A = 0.0f;
        for (int w = 0; w < WPB; ++w) { S += wred[w][0]; A += wred[w][1]; }
        atomicAdd(&acc[0], S);
        atomicAdd(&acc[1], A);
    }
}

__global__ void init_acc(float* acc) {
    if (threadIdx.x < 2) acc[threadIdx.x] = 0.0f;
}

__global__ void finalize_k(const float* __restrict__ acc, float* __restrict__ out) {
    if (threadIdx.x == 0 && blockIdx.x == 0) {
        float s = acc[1];
        int k = 0;
        while (s > 1.0f) { s *= 0.5f; ++k; }   // exact fp32 halving, matches reference
        out[0] = ldexpf(acc[0], -k);           // sum(h) * 2^-k, exact scale
    }
}

extern "C" void kernel_launch(void* const* d_in, const int* in_sizes, int n_in,
                              void* d_out, int out_size, void* d_ws, size_t ws_size,
                              hipStream_t stream) {
    const float* X  = (const float*)d_in[0];
    const float* W  = (const float*)d_in[1];
    const float* b  = (const float*)d_in[2];
    const float* Wi = (const float*)d_in[3];
    float* out = (float*)d_out;
    float* acc = (float*)d_ws;                 // acc[0]=sum, acc[1]=abs-sum

    const int rows  = in_sizes[0] / DD;        // 1,000,000
    const int tiles = rows / 16;               // 62,500 (exact)

    init_acc<<<1, 32, 0, stream>>>(acc);
    mlp_chain_reduce<<<GRID, BLOCK, 0, stream>>>(X, W, b, Wi, acc, tiles);
    finalize_k<<<1, 1, 0, stream>>>(acc, out);
}